// ListMLELoss_39350490366201
// MI455X (gfx1250) — compile-verified
//
#include <hip/hip_runtime.h>
#include <math.h>

#define NN   2048          // candidates per row (fixed by reference)
#define NTH  256           // threads per block = 8 wave32 waves
#define EPERT (NN / NTH)   // 8 elements per thread
#define NEG_INF (-__builtin_inff())

// ---------------------------------------------------------------------------
// CDNA5 async global->LDS staging (ASYNCcnt path). Guarded so a missing/renamed
// builtin falls back to a plain coalesced copy and still compiles.
// Probe round 1 revealed the parameter type: int4* (AS1 global), int4* (AS3 LDS).
// ---------------------------------------------------------------------------
#if defined(__has_builtin)
#if __has_builtin(__builtin_amdgcn_global_load_async_to_lds_b128)
#define USE_ASYNC_LDS 1
#endif
#endif

#ifdef USE_ASYNC_LDS
typedef int v4i __attribute__((ext_vector_type(4)));
typedef __attribute__((address_space(1))) v4i* g_v4i_ptr;
typedef __attribute__((address_space(3))) v4i* l_v4i_ptr;
__device__ __forceinline__ void async_cp16(void* lds, const void* g) {
    __builtin_amdgcn_global_load_async_to_lds_b128(
        (g_v4i_ptr)(const_cast<void*>(g)),
        (l_v4i_ptr)(lds),
        /*imm_offset=*/0, /*cpol=*/0);
}
#endif

// logsumexp of two values; handles -inf identity
__device__ __forceinline__ float lse2(float a, float b) {
    float m = fmaxf(a, b);
    if (m == NEG_INF) return NEG_INF;
    return m + logf(expf(a - m) + expf(b - m));
}

// ---------------------------------------------------------------------------
// Kernel 1: one workgroup per row.
//   LDS bitonic sort (label desc, score carried) + suffix cum-logsumexp scan.
//   Writes per-row loss into row_loss[row].
// ---------------------------------------------------------------------------
__global__ __launch_bounds__(NTH)
void listmle_row_kernel(const float* __restrict__ scores,
                        const float* __restrict__ labels,
                        float* __restrict__ row_loss) {
    __shared__ __align__(16) float s_key[NN];   // labels (sort key, descending)
    __shared__ __align__(16) float s_val[NN];   // scores (payload)
    __shared__ float s_scan[NTH];               // chunk LSE totals / suffix scan
    __shared__ float s_red[NTH];                // loss reduction
    __shared__ float s_cnt[NTH];                // mask-count reduction

    const int tid = threadIdx.x;
    const int row = blockIdx.x;
    const float* gk = labels + (size_t)row * NN;
    const float* gv = scores + (size_t)row * NN;

    // ---- stage row into LDS ----
#ifdef USE_ASYNC_LDS
    // 2048 floats per array / (256 lanes * 4 floats) = 2 b128 transfers each
    for (int p = 0; p < NN; p += NTH * 4) {
        int i = p + tid * 4;
        async_cp16(&s_key[i], gk + i);
        async_cp16(&s_val[i], gv + i);
    }
    asm volatile("s_wait_asynccnt 0" ::: "memory");
    __syncthreads();
#else
    for (int i = tid; i < NN; i += NTH) {
        s_key[i] = gk[i];
        s_val[i] = gv[i];
    }
    __syncthreads();
#endif

    // ---- bitonic sort: key descending, val follows (TEMPERATURE == 1) ----
    for (int k = 2; k <= NN; k <<= 1) {
        for (int j = k >> 1; j > 0; j >>= 1) {
            #pragma unroll
            for (int p = 0; p < EPERT; ++p) {
                int i   = p * NTH + tid;
                int ixj = i ^ j;
                if (ixj > i) {
                    float ki = s_key[i], kj = s_key[ixj];
                    bool up = ((i & k) == 0);          // up => descending run
                    if ((ki < kj) == up) {
                        s_key[i] = kj; s_key[ixj] = ki;
                        float vi = s_val[i];
                        s_val[i] = s_val[ixj]; s_val[ixj] = vi;
                    }
                }
            }
            __syncthreads();
        }
    }

    // ---- chunk-local LSE totals (thread t owns [t*8, t*8+8)) ----
    const int base = tid * EPERT;
    float tot = NEG_INF;
    #pragma unroll
    for (int e = 0; e < EPERT; ++e) tot = lse2(tot, s_val[base + e]);
    s_scan[tid] = tot;
    __syncthreads();

    // ---- Hillis-Steele inclusive *suffix* scan over 256 chunk totals ----
    for (int off = 1; off < NTH; off <<= 1) {
        float v = s_scan[tid];
        float w = (tid + off < NTH) ? s_scan[tid + off] : NEG_INF;
        __syncthreads();
        s_scan[tid] = lse2(v, w);
        __syncthreads();
    }
    float suf = (tid + 1 < NTH) ? s_scan[tid + 1] : NEG_INF;  // exclusive suffix

    // ---- walk own chunk back-to-front: exact rev_lse, accumulate loss ----
    float acc = 0.0f;   // sum of (rev_lse - ss) * mask == -sum(cll * mask)
    float cnt = 0.0f;   // sum of mask (all positions)
    #pragma unroll
    for (int e = EPERT - 1; e >= 0; --e) {
        int idx  = base + e;
        float sc = s_val[idx];
        suf = lse2(suf, sc);                       // rev_lse[idx]
        float m = (s_key[idx] > 0.0f) ? 1.0f : 0.0f;
        cnt += m;
        if (idx < NN - 1) acc += (suf - sc) * m;   // last position contributes 0
    }

    // ---- block reduction -> per-row loss ----
    s_red[tid] = acc;
    s_cnt[tid] = cnt;
    __syncthreads();
    for (int s = NTH >> 1; s > 0; s >>= 1) {
        if (tid < s) {
            s_red[tid] += s_red[tid + s];
            s_cnt[tid] += s_cnt[tid + s];
        }
        __syncthreads();
    }
    if (tid == 0) row_loss[row] = s_red[0] / (s_cnt[0] + 1e-10f);
}

// ---------------------------------------------------------------------------
// Kernel 2: deterministic mean over B row losses.
// ---------------------------------------------------------------------------
__global__ __launch_bounds__(NTH)
void listmle_mean_kernel(const float* __restrict__ row_loss,
                         float* __restrict__ out, int B) {
    __shared__ float sm[NTH];
    float a = 0.0f;
    for (int i = threadIdx.x; i < B; i += NTH) a += row_loss[i];
    sm[threadIdx.x] = a;
    __syncthreads();
    for (int s = NTH >> 1; s > 0; s >>= 1) {
        if (threadIdx.x < s) sm[threadIdx.x] += sm[threadIdx.x + s];
        __syncthreads();
    }
    if (threadIdx.x == 0) out[0] = sm[0] / (float)B;
}

extern "C" void kernel_launch(void* const* d_in, const int* in_sizes, int n_in,
                              void* d_out, int out_size, void* d_ws, size_t ws_size,
                              hipStream_t stream) {
    (void)n_in; (void)out_size; (void)ws_size;
    const float* scores = (const float*)d_in[0];
    const float* labels = (const float*)d_in[1];
    float* out = (float*)d_out;
    float* row_loss = (float*)d_ws;            // B floats of scratch

    const int B = in_sizes[0] / NN;            // 8192 for the reference shapes

    listmle_row_kernel<<<B, NTH, 0, stream>>>(scores, labels, row_loss);
    listmle_mean_kernel<<<1, NTH, 0, stream>>>(row_loss, out, B);
}